// CoralLossV2_13640816132488
// MI455X (gfx1250) — compile-verified
//
#include <hip/hip_runtime.h>
#include <cmath>

typedef float v2f __attribute__((ext_vector_type(2)));
typedef float v8f __attribute__((ext_vector_type(8)));

__device__ __forceinline__ float bce_logits(float x, float z) {
    // max(x,0) - x*z + log1p(exp(-|x|))  (numerically stable BCE-with-logits)
    return fmaxf(x, 0.0f) - x * z + log1pf(__expf(-fabsf(x)));
}

__global__ __launch_bounds__(256) void coral_loss_kernel(
    const float* __restrict__ logits,
    const int*  __restrict__ targets,
    float* __restrict__ out,
    int n4, int kqShift, int kqMask, float invN)
{
    __shared__ float waveSums[8];

    const int tid    = blockIdx.x * blockDim.x + threadIdx.x;
    const int stride = gridDim.x * blockDim.x;

    const float4* __restrict__ lg4 = reinterpret_cast<const float4*>(logits);

    float acc = 0.0f;
    for (int idx = tid; idx < n4; idx += stride) {
        // Prefetch the next grid-stride chunk (speculative, OOB-safe).
        __builtin_prefetch(&lg4[idx + stride], 0, 0);

        float4 x  = lg4[idx];                 // 128-bit coalesced load
        int    row = idx >> kqShift;          // row in (B, K-1)
        int    k0  = (idx & kqMask) << 2;     // first threshold index of this float4
        int    t   = targets[row];            // broadcast within wave, cache-resident

        float z0 = (t > (k0 + 0)) ? 1.0f : 0.0f;
        float z1 = (t > (k0 + 1)) ? 1.0f : 0.0f;
        float z2 = (t > (k0 + 2)) ? 1.0f : 0.0f;
        float z3 = (t > (k0 + 3)) ? 1.0f : 0.0f;

        acc += (bce_logits(x.x, z0) + bce_logits(x.y, z1)) +
               (bce_logits(x.z, z2) + bce_logits(x.w, z3));
    }

    // ---- Exact f32 cross-lane wave reduction via V_WMMA_F32_16X16X4_F32 ----
    // A (16x4): VGPR0 = acc (K=0 lanes 0-15 / K=2 lanes 16-31), VGPR1 = 0.
    // B (4x16): all ones.  => D[m][n] = acc[m] + acc[m+16] for every n.
    // Lane L<16 holds D[0..7][L] in d[0..7]; lane L>=16 holds D[8..15][L-16].
    v2f a; a[0] = acc;  a[1] = 0.0f;
    v2f b; b[0] = 1.0f; b[1] = 1.0f;
    v8f c = {};
    v8f d = __builtin_amdgcn_wmma_f32_16x16x4_f32(
        /*neg_a=*/false, a, /*neg_b=*/false, b,
        /*c_mod=*/(short)0, c, /*reuse_a=*/false, /*reuse_b=*/false);

    float s = ((d[0] + d[1]) + (d[2] + d[3])) + ((d[4] + d[5]) + (d[6] + d[7]));
    // Lanes 0-15 have sum over m=0..7, lanes 16-31 over m=8..15: combine halves.
    s += __shfl_xor(s, 16, 32);

    const int lane = threadIdx.x & 31;
    const int wave = threadIdx.x >> 5;
    if (lane == 0) waveSums[wave] = s;
    __syncthreads();

    if (threadIdx.x == 0) {
        const int nw = blockDim.x >> 5;
        float blockSum = 0.0f;
        for (int w = 0; w < nw; ++w) blockSum += waveSums[w];
        atomicAdd(out, blockSum * invN);   // fold mean scale into the atomic
    }
}

extern "C" void kernel_launch(void* const* d_in, const int* in_sizes, int n_in,
                              void* d_out, int out_size, void* d_ws, size_t ws_size,
                              hipStream_t stream) {
    const float* logits  = (const float*)d_in[0];
    const int*   targets = (const int*)d_in[1];
    float*       out     = (float*)d_out;

    const long long nElem = (long long)in_sizes[0];   // B * (K-1) = 33,554,432
    const int B   = in_sizes[1];                      // 524288
    const int km1 = (int)(nElem / (long long)B);      // 64
    const int kq  = km1 >> 2;                         // float4s per row = 16 (pow2)
    int shift = 0;
    while ((1 << shift) < kq) ++shift;                // 4
    const int   mask = kq - 1;
    const int   n4   = (int)(nElem >> 2);             // 8,388,608 float4s
    const float invN = 1.0f / (float)nElem;

    // Zero the scalar accumulator (graph-capture safe).
    hipMemsetAsync(d_out, 0, (size_t)out_size * sizeof(float), stream);

    // 2048 blocks x 256 threads = 16384 wave32s: enough latency-hiding to
    // saturate 23.3 TB/s; each thread handles 16 float4s.
    const int threads = 256;
    const int blocks  = 2048;
    coral_loss_kernel<<<blocks, threads, 0, stream>>>(
        logits, targets, out, n4, shift, mask, invN);
}